// V2S_CIS_89902255440906
// MI455X (gfx1250) — compile-verified
//
#include <hip/hip_runtime.h>
#include <hip/hip_bf16.h>
#include <math.h>

// Problem constants
#define Bn 64
#define Sn 20
#define Wn 1024
#define NCn 80
#define DLn 300
#define DEn 128
#define KTOT (Wn * DEn)   // 131072
#define KSPLIT 64
#define KCHUNK (KTOT / KSPLIT) // 2048

typedef float v2f __attribute__((ext_vector_type(2)));
typedef float v4f __attribute__((ext_vector_type(4)));
typedef float v8f __attribute__((ext_vector_type(8)));

// leaky_relu(x, 0.01) == max(x, 0.01*x) for all finite x -> single v_pk_max_num_f32
__device__ __forceinline__ float lrelu(float x) { return fmaxf(x, 0.01f * x); }

// ---------------------------------------------------------------------------
// Kernel 1: per-batch setup. argmax ids -> multi-hot mask -> weight_cls
// softmax -> weight_pri -> A0/A1 closed form -> word_prob gather.
// One block per batch sample, 256 threads.
// ---------------------------------------------------------------------------
__global__ __launch_bounds__(256) void k_setup(
    const float* __restrict__ word_recon, const float* __restrict__ beta_ptr,
    const float* __restrict__ metrix_sl, const float* __restrict__ cls_word_prob,
    const float* __restrict__ w_sl, const float* __restrict__ b_sl,
    const int* __restrict__ label, const int* __restrict__ kptr,
    float* __restrict__ wp_ws, float* __restrict__ A0, float* __restrict__ A1,
    float* __restrict__ out_wp, float* __restrict__ out_wordprob)
{
    const int b = blockIdx.x;
    const int t = threadIdx.x;
    __shared__ float red_v[256];
    __shared__ int   red_i[256];
    __shared__ unsigned mask[32];
    __shared__ float wcls[NCn];
    __shared__ float wp_s[Wn];
    __shared__ float scal[3];   // Wtot, Wsel, nsel

    if (t < 32) mask[t] = 0u;
    __syncthreads();

    const float beta = beta_ptr[0];
    int kk = kptr[0]; if (kk > Sn) kk = Sn;

    // --- argmax per timestep, first-index tie rule (matches jnp.argmax) ---
    for (int s = 0; s < kk; ++s) {
        const float* row = word_recon + ((size_t)b * Sn + s) * Wn;
        float bv = -INFINITY; int bi = 0;
        for (int j = t; j < Wn; j += 256) {
            float v = row[j];
            if (v > bv) { bv = v; bi = j; }
        }
        red_v[t] = bv; red_i[t] = bi;
        __syncthreads();
        for (int off = 128; off > 0; off >>= 1) {
            if (t < off) {
                float v2 = red_v[t + off]; int i2 = red_i[t + off];
                if (v2 > red_v[t] || (v2 == red_v[t] && i2 < red_i[t])) {
                    red_v[t] = v2; red_i[t] = i2;
                }
            }
            __syncthreads();
        }
        if (t == 0) { int id = red_i[0]; atomicOr(&mask[id >> 5], 1u << (id & 31)); }
        __syncthreads();
    }

    // --- weight_cls = softmax(mh @ w_sl.T + b_sl) : sum selected columns ---
    if (t < NCn) {
        float acc = b_sl[t];
        for (int wd = 0; wd < 32; ++wd) {
            unsigned m = mask[wd];
            while (m) {
                int bit = __ffs(m) - 1; m &= m - 1;
                acc += w_sl[t * Wn + wd * 32 + bit];
            }
        }
        wcls[t] = acc;
    }
    __syncthreads();
    if (t == 0) {
        float mx = wcls[0];
        for (int c = 1; c < NCn; ++c) mx = fmaxf(mx, wcls[c]);
        float sm = 0.0f;
        for (int c = 0; c < NCn; ++c) { float e = __expf(wcls[c] - mx); wcls[c] = e; sm += e; }
        float inv = 1.0f / sm;
        for (int c = 0; c < NCn; ++c) wcls[c] *= inv;
    }
    __syncthreads();

    // --- weight_pri = weight_cls @ metrix_sl ---
    for (int i = t; i < Wn; i += 256) {
        float acc = 0.0f;
        for (int c = 0; c < NCn; ++c) acc += wcls[c] * metrix_sl[c * Wn + i];
        wp_s[i] = acc;
        wp_ws[b * Wn + i]  = acc;
        out_wp[b * Wn + i] = acc;
    }
    __syncthreads();

    // --- Wtot / Wsel reductions + nsel popcount ---
    float pt = 0.0f, ps = 0.0f;
    for (int i = t; i < Wn; i += 256) {
        float v = wp_s[i];
        pt += v;
        if ((mask[i >> 5] >> (i & 31)) & 1u) ps += v;
    }
    red_v[t] = pt; __syncthreads();
    for (int off = 128; off > 0; off >>= 1) { if (t < off) red_v[t] += red_v[t + off]; __syncthreads(); }
    if (t == 0) scal[0] = red_v[0];
    __syncthreads();
    red_v[t] = ps; __syncthreads();
    for (int off = 128; off > 0; off >>= 1) { if (t < off) red_v[t] += red_v[t + off]; __syncthreads(); }
    if (t == 0) {
        scal[1] = red_v[0];
        int n = 0;
        for (int wd = 0; wd < 32; ++wd) n += __popc(mask[wd]);
        scal[2] = (float)n;
    }
    __syncthreads();

    const float Wtot = scal[0], Wsel = scal[1], nsel = scal[2];
    const float omb = 1.0f - beta;
    for (int i = t; i < Wn; i += 256) {
        float w = wp_s[i];
        bool sel = (mask[i >> 5] >> (i & 31)) & 1u;
        A1[b * Wn + i] = omb * (sel ? nsel : 0.0f) + beta * w * Wsel;
        A0[b * Wn + i] = omb * (sel ? 0.0f : 1.0f) + beta * w * (Wtot - Wsel);
    }

    // --- word_prob = cls_word_prob[label[b]] ---
    const int lb = label[b];
    for (int i = t; i < Wn; i += 256)
        out_wordprob[b * Wn + i] = cls_word_prob[lb * Wn + i];
}

// ---------------------------------------------------------------------------
// Kernel 2: s0/s1 = emb_table[{0,1}] @ w_lw.T + b_lw ; also zero accum.
// ---------------------------------------------------------------------------
__global__ __launch_bounds__(128) void k_s0s1(
    const float* __restrict__ emb, const float* __restrict__ w_lw,
    const float* __restrict__ b_lw, float* __restrict__ s0,
    float* __restrict__ s1, float* __restrict__ accum)
{
    const int d = threadIdx.x;
    float acc0 = 0.0f, acc1 = 0.0f;
    for (int l = 0; l < DLn; ++l) {
        float wv = w_lw[d * DLn + l];
        acc0 += emb[l] * wv;
        acc1 += emb[DLn + l] * wv;
    }
    s0[d] = acc0 + b_lw[d];
    s1[d] = acc1 + b_lw[d];
    for (int i = d; i < Bn * DEn; i += 128) accum[i] = 0.0f;
}

// ---------------------------------------------------------------------------
// Kernel 3: adj_pri[b,i,j] = wp[b,i]*wp[b,j]. 256 MB of pure streaming
// stores (exceeds 192 MB L2) -> float4 non-temporal stores, one block per
// (b,i) row; each wave stores 512 contiguous bytes.
// ---------------------------------------------------------------------------
__global__ __launch_bounds__(256) void k_adjpri(
    const float* __restrict__ wp, float* __restrict__ out)
{
    const int row = blockIdx.x;          // b*W + i
    const int b = row >> 10;
    const float wpi = wp[row];
    const v4f* src = (const v4f*)(wp + b * Wn);
    v4f* dst = (v4f*)(out + (size_t)row * Wn);
    v4f v = src[threadIdx.x];
    v4f r = { wpi * v.x, wpi * v.y, wpi * v.z, wpi * v.w };
    __builtin_nontemporal_store(r, &dst[threadIdx.x]);
}

// ---------------------------------------------------------------------------
// Kernel 4: WMMA GEMM. out[64,128] += feat[64,131072] @ w_gcn.T[131072,128]
// where feat(b, w*128+e) = lrelu(A1[b,w]*s1[e] + A0[b,w]*s0[e]) generated
// on the fly (never materialized). V_WMMA_F32_16X16X4_F32, one wave per
// block, K-split with f32 global atomics. w_gcn (67 MB) stays L2-resident
// (192 MB) across the 4 M-tile re-reads.
// A frag (16x4 f32, ISA 7.12.2): lanes 0-15 hold K=0,1; lanes 16-31 K=2,3;
// M = lane&15. B frag mirrored over N. C/D: 8 VGPRs per ISA table.
// ---------------------------------------------------------------------------
__global__ __launch_bounds__(32) void k_gemm(
    const float* __restrict__ A0, const float* __restrict__ A1,
    const float* __restrict__ s0, const float* __restrict__ s1,
    const float* __restrict__ w_gcn, float* __restrict__ accum)
{
    const int mtile = blockIdx.x;   // 0..3   (M = 64)
    const int ntile = blockIdx.y;   // 0..7   (N = 128)
    const int kchunk = blockIdx.z;  // 0..KSPLIT-1
    const int lane = threadIdx.x;
    const int m = lane & 15;
    const int half = lane >> 4;
    const int klo = half * 2;

    __shared__ float ls0[DEn];
    __shared__ float ls1[DEn];
    for (int i = lane; i < DEn; i += 32) { ls0[i] = s0[i]; ls1[i] = s1[i]; }
    __syncthreads();

    const int row = mtile * 16 + m;              // batch row of A
    const int ncol = ntile * 16 + m;             // w_gcn row supplying B col
    const float* __restrict__ A0r = A0 + row * Wn;
    const float* __restrict__ A1r = A1 + row * Wn;
    const float* __restrict__ Br  = w_gcn + (size_t)ncol * KTOT;

    const int kbase = kchunk * KCHUNK;
    v8f c = {0.f, 0.f, 0.f, 0.f, 0.f, 0.f, 0.f, 0.f};

    for (int kw = kbase; kw < kbase + KCHUNK; kw += DEn) {
        const int w = kw >> 7;                   // word index, constant for 32 steps
        const float a0v = A0r[w];
        const float a1v = A1r[w];
        #pragma unroll 8
        for (int ke = 0; ke < DEn; ke += 4) {
            const int e = ke + klo;
            float fa = lrelu(a1v * ls1[e]     + a0v * ls0[e]);
            float fb = lrelu(a1v * ls1[e + 1] + a0v * ls0[e + 1]);
            v2f a = { fa, fb };
            v2f bb = *(const v2f*)(Br + kw + ke + klo);
            c = __builtin_amdgcn_wmma_f32_16x16x4_f32(
                    false, a, false, bb, (short)0, c, false, false);
        }
    }

    #pragma unroll
    for (int v = 0; v < 8; ++v) {
        const int gm = mtile * 16 + v + half * 8;
        const int gn = ntile * 16 + m;
        atomicAdd(&accum[gm * DEn + gn], c[v]);
    }
}

// ---------------------------------------------------------------------------
// Kernel 5: feat = lrelu(accum + b_gcn); output = feat @ w_cls.T + b_cls
// ---------------------------------------------------------------------------
__global__ __launch_bounds__(128) void k_final(
    const float* __restrict__ accum, const float* __restrict__ b_gcn,
    const float* __restrict__ w_cls, const float* __restrict__ b_cls,
    float* __restrict__ out_output, float* __restrict__ out_feat)
{
    const int b = blockIdx.x, t = threadIdx.x;
    __shared__ float f[DEn];
    float v = lrelu(accum[b * DEn + t] + b_gcn[t]);
    out_feat[b * DEn + t] = v;
    f[t] = v;
    __syncthreads();
    if (t < NCn) {
        float acc = b_cls[t];
        for (int d = 0; d < DEn; ++d) acc += f[d] * w_cls[t * DEn + d];
        out_output[b * NCn + t] = acc;
    }
}

// ---------------------------------------------------------------------------
extern "C" void kernel_launch(void* const* d_in, const int* in_sizes, int n_in,
                              void* d_out, int out_size, void* d_ws, size_t ws_size,
                              hipStream_t stream) {
    const float* word_recon   = (const float*)d_in[0];
    const float* beta_pri     = (const float*)d_in[1];
    const float* metrix_sl    = (const float*)d_in[2];
    const float* cls_word_prob= (const float*)d_in[3];
    const float* emb_table    = (const float*)d_in[4];
    const float* w_sl         = (const float*)d_in[5];
    const float* b_sl         = (const float*)d_in[6];
    const float* w_lw         = (const float*)d_in[7];
    const float* b_lw         = (const float*)d_in[8];
    const float* w_gcn        = (const float*)d_in[9];
    const float* b_gcn        = (const float*)d_in[10];
    const float* w_cls        = (const float*)d_in[11];
    const float* b_cls        = (const float*)d_in[12];
    const int*   label        = (const int*)d_in[13];
    const int*   kptr         = (const int*)d_in[14];

    float* out = (float*)d_out;
    float* out_output   = out;                                  // [B,NC]
    float* out_feat     = out + Bn * NCn;                       // [B,DE]
    float* out_wp       = out + Bn * NCn + Bn * DEn;            // [B,1,W]
    float* out_adjpri   = out_wp + Bn * Wn;                     // [B,W,W]
    float* out_wordprob = out_adjpri + (size_t)Bn * Wn * Wn;    // [B,1,W]

    float* ws = (float*)d_ws;
    float* wp_ws = ws;                      // [B,W]
    float* A0    = wp_ws + Bn * Wn;         // [B,W]
    float* A1    = A0 + Bn * Wn;            // [B,W]
    float* s0    = A1 + Bn * Wn;            // [DE]
    float* s1    = s0 + DEn;                // [DE]
    float* accum = s1 + DEn;                // [B,DE]

    k_setup<<<Bn, 256, 0, stream>>>(word_recon, beta_pri, metrix_sl, cls_word_prob,
                                    w_sl, b_sl, label, kptr,
                                    wp_ws, A0, A1, out_wp, out_wordprob);
    k_s0s1<<<1, 128, 0, stream>>>(emb_table, w_lw, b_lw, s0, s1, accum);
    k_adjpri<<<Bn * Wn, 256, 0, stream>>>(wp_ws, out_adjpri);
    k_gemm<<<dim3(4, 8, KSPLIT), 32, 0, stream>>>(A0, A1, s0, s1, w_gcn, accum);
    k_final<<<Bn, 128, 0, stream>>>(accum, b_gcn, w_cls, b_cls, out_output, out_feat);
}